// PiecewiseLinearShapeNN2D_29703993819696
// MI455X (gfx1250) — compile-verified
//
#include <hip/hip_runtime.h>

#define NX 1024
#define NY 1024

typedef float v4f __attribute__((ext_vector_type(4)));

// ---------------------------------------------------------------------------
// Kernel 1: build the two adaptive grids (one block per grid, 1024 threads).
// grid = [0, cum[0]/T, ..., cum[1021]/T, 1], cum = cumsum(max(softplus(incr),1e-6))
// ---------------------------------------------------------------------------
__global__ void __launch_bounds__(1024) build_grids_kernel(
        const float* __restrict__ incx,
        const float* __restrict__ incy,
        float* __restrict__ grids)
{
    __shared__ float sA[1024];
    __shared__ float sB[1024];
    const int tid = threadIdx.x;
    const float* incr = (blockIdx.x == 0) ? incx : incy;
    float* gout = grids + blockIdx.x * 1024;

    float v = 0.0f;
    if (tid < 1023) {
        float x  = incr[tid];
        float sp = (x > 0.0f) ? (x + log1pf(expf(-x))) : log1pf(expf(x));
        v = fmaxf(sp, 1e-6f);
    }
    sA[tid] = v;
    __syncthreads();

    // Hillis-Steele inclusive scan over 1024 entries (entry 1023 is 0 pad)
    float* src = sA;
    float* dst = sB;
    for (int off = 1; off < 1024; off <<= 1) {
        float val = src[tid];
        if (tid >= off) val += src[tid - off];
        dst[tid] = val;
        __syncthreads();
        float* tmp = src; src = dst; dst = tmp;
    }
    const float total = src[1022];   // cum[-1] over the 1023 increments

    if (tid == 0) {
        gout[0] = 0.0f;              // masked endpoint = init_grid[0]
    } else if (tid == 1023) {
        gout[1023] = 1.0f;           // masked endpoint = init_grid[-1]
    } else {
        gout[tid] = src[tid - 1] / total;
    }
}

// ---------------------------------------------------------------------------
// Exact searchsorted(side='left') via analytic guess + monotone fix-up walk.
// Correct for any sorted grid; ~1-2 LDS probes for this near-uniform grid
// (vs 10 for a full binary search). Returns interval index in [0, 1022].
// ---------------------------------------------------------------------------
__device__ __forceinline__ int interval_idx(const float* g, float v)
{
    int i = (int)(v * 1023.0f);                 // uniform-grid guess
    i = (i < 0) ? 0 : ((i > 1023) ? 1023 : i);
    while (i < 1023 && g[i] < v) i++;           // enforce g[i] >= v (or i==1023)
    while (i > 0 && g[i - 1] >= v) i--;         // enforce g[i-1] < v
    // i == lower_bound position (capped at 1023, which clamps identically)
    int idx = i - 1;
    idx = (idx < 0) ? 0 : idx;
    return (idx > NX - 2) ? (NX - 2) : idx;
}

__device__ __forceinline__ float eval_point(const float* gx, const float* gy,
                                            const float* __restrict__ u,
                                            float xe, float ye)
{
    const int ix = interval_idx(gx, xe);
    const int iy = interval_idx(gy, ye);
    const float x_i   = gx[ix];
    const float x_ip1 = gx[ix + 1];
    const float y_i   = gy[iy];
    const float y_ip1 = gy[iy + 1];
    const float dx = fmaxf(x_ip1 - x_i, 1e-10f);
    const float dy = fmaxf(y_ip1 - y_i, 1e-10f);
    const float n1x = (x_ip1 - xe) / dx;
    const float n2x = (xe - x_i) / dx;
    const float n1y = (y_ip1 - ye) / dy;
    const float n2y = (ye - y_i) / dy;
    const int base = ix * NY + iy;
    const float u00 = u[base];
    const float u01 = u[base + 1];
    const float u10 = u[base + NY];
    const float u11 = u[base + NY + 1];
    return n1x * n1y * u00 + n2x * n1y * u10 + n1x * n2y * u01 + n2x * n2y * u11;
}

// ---------------------------------------------------------------------------
// Kernel 2: 4 points per thread. x_eval streamed with NT loads (2 x b128),
// output with one NT b128 store, so the 4MB u field stays hot in L2.
// Grids staged into LDS with gfx1250 async-to-LDS (ASYNCcnt) loads.
// ---------------------------------------------------------------------------
__global__ void __launch_bounds__(256) interp_kernel(
        const float* __restrict__ xy,
        const float* __restrict__ u,
        const float* __restrict__ grids,
        float* __restrict__ out,
        int npts)
{
    __shared__ __align__(16) float sg[2048];   // gx[1024] ++ gy[1024]
    const int tid = threadIdx.x;

    // CDNA5 async copy: 2048 floats = 512 x b128 chunks, 256 lanes -> 2 iters.
    for (int i = tid * 4; i < 2048; i += 256 * 4) {
        unsigned lds_off =
            (unsigned)(unsigned long long)(const void*)&sg[i];
        unsigned long long ga =
            (unsigned long long)(const void*)(grids + i);
        asm volatile("global_load_async_to_lds_b128 %0, %1, off"
                     :: "v"(lds_off), "v"(ga)
                     : "memory");
    }
    asm volatile("s_wait_asynccnt 0x0" ::: "memory");
    __syncthreads();

    const float* gx = sg;
    const float* gy = sg + 1024;

    const int quad = blockIdx.x * 256 + tid;   // 4 points per thread
    const int p0 = quad * 4;
    if (p0 >= npts) return;

    if (p0 + 3 < npts) {
        // fast path: 2 NT b128 loads (4 points), 1 NT b128 store
        const v4f a = __builtin_nontemporal_load((const v4f*)(xy + 8 * quad));
        const v4f b = __builtin_nontemporal_load((const v4f*)(xy + 8 * quad + 4));
        v4f r;
        r.x = eval_point(gx, gy, u, a.x, a.y);
        r.y = eval_point(gx, gy, u, a.z, a.w);
        r.z = eval_point(gx, gy, u, b.x, b.y);
        r.w = eval_point(gx, gy, u, b.z, b.w);
        __builtin_nontemporal_store(r, (v4f*)(out + p0));
    } else {
        // tail: scalar per point
        for (int p = p0; p < npts; ++p) {
            const float xe = xy[2 * p];
            const float ye = xy[2 * p + 1];
            out[p] = eval_point(gx, gy, u, xe, ye);
        }
    }
}

// ---------------------------------------------------------------------------
// Inputs: x_eval (8M x 2 f32), increments_x (1023 f32), increments_y (1023
// f32), u (1024x1024 f32). Output: 8M f32. d_ws: 8 KB grid scratch.
// ---------------------------------------------------------------------------
extern "C" void kernel_launch(void* const* d_in, const int* in_sizes, int n_in,
                              void* d_out, int out_size, void* d_ws, size_t ws_size,
                              hipStream_t stream)
{
    const float* x_eval = (const float*)d_in[0];
    const float* incx   = (const float*)d_in[1];
    const float* incy   = (const float*)d_in[2];
    const float* u      = (const float*)d_in[3];
    float* out   = (float*)d_out;
    float* grids = (float*)d_ws;
    (void)in_sizes; (void)n_in; (void)ws_size;

    build_grids_kernel<<<2, 1024, 0, stream>>>(incx, incy, grids);

    const int npts    = out_size;
    const int n_quads = (npts + 3) / 4;
    const int blocks  = (n_quads + 255) / 256;
    interp_kernel<<<blocks, 256, 0, stream>>>(x_eval, u, grids, out, npts);
}